// AllPassFilterAugmentation_88287347736693
// MI455X (gfx1250) — compile-verified
//
#include <hip/hip_runtime.h>
#include <math.h>

typedef __attribute__((ext_vector_type(2))) float v2f;
typedef __attribute__((ext_vector_type(8))) float v8f;

#define ROWS            64
#define TLEN            480000
#define TILE            256
#define TILES_PER_ROW   (TLEN / TILE)     // 1875
#define TILES_PER_WAVE  4
#define WAVES_PER_BLOCK 8
#define NTAPS           48                 // pole radius ~0.414 -> g[48] ~ 1e-18

// y[n] = sum_k g[k] x[n-k]  (truncated impulse response of the allpass biquad,
// exact to below f32 epsilon) computed as banded-Toeplitz matmul on
// V_WMMA_F32_16X16X4_F32. Each wave produces a 16x16 tile of 256 consecutive
// outputs: Y = sum_{b=0..2} X_b * G_b, 12 WMMAs per tile.
__global__ __launch_bounds__(256) void allpass_fir_wmma(const float* __restrict__ x,
                                                        float* __restrict__ y) {
    // Zero-padded impulse response table: [0..47]=0 (negative Toeplitz idx), [48..95]=g[0..47]
    __shared__ float gtab[96];
    const int tid = threadIdx.x;
    if (tid < 48) gtab[tid] = 0.0f;
    if (tid == 0) {
        // allpass_biquad coefficients (sr=16000, f0=4000, Q=0.707), normalized by a0
        double w0    = 6.283185307179586 * 4000.0 / 16000.0;
        double alpha = sin(w0) / (2.0 * 0.707);
        double cw    = cos(w0);
        double a0    = 1.0 + alpha;
        float b0 = (float)((1.0 - alpha) / a0);
        float b1 = (float)((-2.0 * cw) / a0);
        float b2 = (float)((1.0 + alpha) / a0);
        float a1 = b1, a2 = b0;           // allpass: denominator = reversed numerator
        float y1 = 0.f, y2 = 0.f, x1 = 0.f, x2 = 0.f;
        for (int n = 0; n < NTAPS; ++n) {
            float xn = (n == 0) ? 1.0f : 0.0f;
            float yn = b0 * xn + b1 * x1 + b2 * x2 - a1 * y1 - a2 * y2;
            gtab[48 + n] = yn;
            x2 = x1; x1 = xn; y2 = y1; y1 = yn;
        }
    }
    __syncthreads();

    const int lane = tid & 31;
    const int wave = tid >> 5;
    const int h    = lane >> 4;   // wave half (selects K pair for A/B operands)
    const int j    = lane & 15;   // A: M row index; B/D: N column index

    // Constant B operands: G_b[c][jcol] = g[16b + jcol - c], 3 Toeplitz blocks x 4 K-slices.
    // B layout for 16x16x4: lanes 0-15 hold rows {0,1}, lanes 16-31 rows {2,3}.
    v2f Bv[12];
#pragma unroll
    for (int b = 0; b < 3; ++b) {
#pragma unroll
        for (int q = 0; q < 4; ++q) {
            int row  = 4 * q + 2 * h;
            int base = 48 + 16 * b + j;
            v2f t;
            t.x = gtab[base - row];
            t.y = gtab[base - row - 1];
            Bv[b * 4 + q] = t;
        }
    }

    // Tile assignment: row/tpos maintained incrementally (no per-iteration 64-bit div).
    const long tile0 = ((long)blockIdx.x * WAVES_PER_BLOCK + wave) * TILES_PER_WAVE;
    int row  = (int)(tile0 / TILES_PER_ROW);
    int tpos = (int)(tile0 % TILES_PER_ROW);

#pragma unroll 1
    for (int i = 0; i < TILES_PER_WAVE; ++i) {
        long rbase = (long)row * TLEN;
        int  t0    = tpos * TILE;
        const float* xr = x + rbase;

        // Dual accumulators: split the 12-WMMA RAW chain into two chains of 6.
        v8f acc0 = {};
        v8f acc1 = {};
        if (t0 >= 32) {
            // Interior tile: aligned v2f loads of shifted input tiles (A operands).
#pragma unroll
            for (int p = 0; p < 12; ++p) {
                int b = p >> 2, q = p & 3;
                int idx = t0 - 16 * b + 16 * j + 4 * q + 2 * h;  // even -> 8B aligned
                v2f a = *(const v2f*)(xr + idx);
                if (p & 1)
                    acc1 = __builtin_amdgcn_wmma_f32_16x16x4_f32(
                        false, a, false, Bv[p], (short)0, acc1, false, false);
                else
                    acc0 = __builtin_amdgcn_wmma_f32_16x16x4_f32(
                        false, a, false, Bv[p], (short)0, acc0, false, false);
            }
        } else {
            // Row head (wave-uniform branch): zero history -> filter starts from rest.
#pragma unroll
            for (int p = 0; p < 12; ++p) {
                int b = p >> 2, q = p & 3;
                int idx = t0 - 16 * b + 16 * j + 4 * q + 2 * h;
                float v0 = xr[idx >= 0 ? idx : 0];
                float v1 = xr[(idx + 1) >= 0 ? (idx + 1) : 0];
                v2f a;
                a.x = (idx >= 0) ? v0 : 0.0f;
                a.y = ((idx + 1) >= 0) ? v1 : 0.0f;
                if (p & 1)
                    acc1 = __builtin_amdgcn_wmma_f32_16x16x4_f32(
                        false, a, false, Bv[p], (short)0, acc1, false, false);
                else
                    acc0 = __builtin_amdgcn_wmma_f32_16x16x4_f32(
                        false, a, false, Bv[p], (short)0, acc0, false, false);
            }
        }

        // Merge accumulators and store. D layout: VGPR v, lane -> (M = v + 8h, N = j).
        float* yr = y + rbase + t0;
#pragma unroll
        for (int v = 0; v < 8; ++v) {
            yr[16 * (v + 8 * h) + j] = acc0[v] + acc1[v];
        }

        if (++tpos == TILES_PER_ROW) { tpos = 0; ++row; }
    }
}

extern "C" void kernel_launch(void* const* d_in, const int* in_sizes, int n_in,
                              void* d_out, int out_size, void* d_ws, size_t ws_size,
                              hipStream_t stream) {
    (void)in_sizes; (void)n_in; (void)d_ws; (void)ws_size; (void)out_size;
    const float* x = (const float*)d_in[0];
    float*       y = (float*)d_out;
    const int total_tiles = ROWS * TILES_PER_ROW;                        // 120000
    const int blocks = total_tiles / (WAVES_PER_BLOCK * TILES_PER_WAVE); // 3750 exact
    allpass_fir_wmma<<<blocks, WAVES_PER_BLOCK * 32, 0, stream>>>(x, y);
}